// Wasserstein_POT_69320772157807
// MI455X (gfx1250) — compile-verified
//
#include <hip/hip_runtime.h>
#include <float.h>

typedef __attribute__((ext_vector_type(2))) float v2f;
typedef __attribute__((ext_vector_type(8))) float v8f;

#define T_DIM 2048
#define R_DIM 256
#define S_DIM 8
#define B_DIM 4
#define BS_DIM 32           // B*S
#define NPROB 8192          // B*S*R
#define TSPLIT 16
#define TCHUNK 128          // T_DIM / TSPLIT
#define WPB 4               // waves (=problems) per block in k_w2

#if defined(__has_builtin)
#  if __has_builtin(__builtin_amdgcn_wmma_f32_16x16x4_f32)
#    define HAVE_WMMA_F32X4 1
#  endif
#endif

// Order-preserving float<->uint map so atomicMin(uint) == float min (deterministic).
static __device__ __forceinline__ unsigned mapMin(float f) {
  unsigned u = __float_as_uint(f);
  return u ^ ((u >> 31) ? 0xFFFFFFFFu : 0x80000000u);
}
static __device__ __forceinline__ float unmapMin(unsigned u) {
  unsigned b = (u & 0x80000000u) ? (u ^ 0x80000000u) : ~u;
  return __uint_as_float(b);
}

__global__ void k_init(unsigned* minslot) { minslot[0] = 0xFFFFFFFFu; }

// One block per (bs, tchunk); thread = r => fully coalesced streaming.
// Produces per-problem partial sums (raw, unshifted) and the global raw min.
__global__ __launch_bounds__(256) void k_partial(const float* __restrict__ x,
                                                 const float* __restrict__ y,
                                                 float* __restrict__ pu,
                                                 float* __restrict__ pv,
                                                 unsigned* __restrict__ minslot) {
  const int blk = blockIdx.x;  // bs*TSPLIT + tc
  const int r = threadIdx.x;
  const size_t base = (size_t)blk * TCHUNK * R_DIM + r;  // == (bs*T + tc*TCHUNK)*R + r
  const float* gx = x + base;
  const float* gy = y + base;
  float su = 0.f, sv = 0.f, mn = FLT_MAX;
  for (int t = 0; t < TCHUNK; ++t) {
    __builtin_prefetch(gx + (size_t)(t + 8) * R_DIM, 0, 1);
    __builtin_prefetch(gy + (size_t)(t + 8) * R_DIM, 0, 1);
    float a = gx[(size_t)t * R_DIM];
    float b = gy[(size_t)t * R_DIM];
    su += a;
    sv += b;
    mn = fminf(mn, fminf(a, b));
  }
  pu[blk * R_DIM + r] = su;
  pv[blk * R_DIM + r] = sv;
  __shared__ float red[256];
  red[r] = mn;
  __syncthreads();
  for (int s = 128; s > 0; s >>= 1) {
    if (r < s) red[r] = fminf(red[r], red[r + s]);
    __syncthreads();
  }
  if (r == 0) atomicMin(minslot, mapMin(red[0]));
}

// Fixed-order combine of TSPLIT partials per problem (deterministic).
__global__ __launch_bounds__(256) void k_sums(const float* __restrict__ pu,
                                              const float* __restrict__ pv,
                                              float* __restrict__ su,
                                              float* __restrict__ sv) {
  const int p = blockIdx.x * 256 + threadIdx.x;
  const int bs = p >> 8;
  const int r = p & 255;
  float a = 0.f, b = 0.f;
  for (int tc = 0; tc < TSPLIT; ++tc) {
    a += pu[(bs * TSPLIT + tc) * R_DIM + r];
    b += pv[(bs * TSPLIT + tc) * R_DIM + r];
  }
  su[p] = a;
  sv[p] = b;
}

// One wave per problem. Build normalized CDFs in LDS via wave32 scans,
// then merge-path partition + 128 merge steps per lane.
__global__ __launch_bounds__(128) void k_w2(const float* __restrict__ x,
                                            const float* __restrict__ y,
                                            const unsigned* __restrict__ minslot,
                                            const float* __restrict__ sums_u,
                                            const float* __restrict__ sums_v,
                                            float* __restrict__ w2out) {
  __shared__ float U[WPB][T_DIM];  // 32 KB
  __shared__ float V[WPB][T_DIM];  // 32 KB
  const int wave = threadIdx.x >> 5;
  const int lane = threadIdx.x & 31;
  const int p = blockIdx.x * WPB + wave;
  const size_t base = (size_t)(p >> 8) * T_DIM * R_DIM + (size_t)(p & 255);
  const float* gx = x + base;
  const float* gy = y + base;

  const float mn = unmapMin(minslot[0]);
  const float shift = 1.1f * fminf(mn, 0.0f);  // u[t] = x[t] - shift
  const float inv_su = 1.0f / (sums_u[p] - (float)T_DIM * shift);
  const float inv_sv = 1.0f / (sums_v[p] - (float)T_DIM * shift);
  float* Up = U[wave];
  float* Vp = V[wave];

  // --- normalized CDF of u into LDS (32 elements per round, carry chain) ---
  float carry = 0.f;
  for (int m = 0; m < T_DIM / 32; ++m) {
    int t = m * 32 + lane;
    __builtin_prefetch(gx + (size_t)(t + 64) * R_DIM, 0, 1);
    float sc = gx[(size_t)t * R_DIM] - shift;
#pragma unroll
    for (int off = 1; off < 32; off <<= 1) {
      float o = __shfl_up(sc, off, 32);
      if (lane >= off) sc += o;
    }
    Up[t] = (carry + sc) * inv_su;
    carry += __shfl(sc, 31, 32);
  }
  // --- normalized CDF of v ---
  carry = 0.f;
  for (int m = 0; m < T_DIM / 32; ++m) {
    int t = m * 32 + lane;
    __builtin_prefetch(gy + (size_t)(t + 64) * R_DIM, 0, 1);
    float sc = gy[(size_t)t * R_DIM] - shift;
#pragma unroll
    for (int off = 1; off < 32; off <<= 1) {
      float o = __shfl_up(sc, off, 32);
      if (lane >= off) sc += o;
    }
    Vp[t] = (carry + sc) * inv_sv;
    carry += __shfl(sc, 31, 32);
  }
  __syncthreads();

  // --- merge-path: lane covers merged indices [k0, k0+128) ---
  const int n = T_DIM;
  const int k0 = lane * (2 * T_DIM / 32);
  int lo = k0 - n;
  if (lo < 0) lo = 0;
  int hi = (k0 < n) ? k0 : n;
  while (lo < hi) {  // smallest i with V[k0-i-1] < U[i]  (ties -> u first)
    int mid = (lo + hi) >> 1;
    int vj = k0 - mid - 1;
    bool pred = (vj < 0) || (Vp[vj] < Up[mid]);
    if (pred) hi = mid;
    else lo = mid + 1;
  }
  int i = lo, j = k0 - lo;
  float prev = 0.f;
  if (k0 > 0) {
    float a1 = (i > 0) ? Up[i - 1] : -FLT_MAX;
    float b1 = (j > 0) ? Vp[j - 1] : -FLT_MAX;
    prev = fmaxf(a1, b1);
  }
  const float invn = 1.0f / (float)T_DIM;
  float acc = 0.f;
  float ucur = (i < n) ? Up[i] : FLT_MAX;
  float vcur = (j < n) ? Vp[j] : FLT_MAX;
  for (int s = 0; s < 2 * T_DIM / 32; ++s) {
    bool takeU = (ucur <= vcur);
    float q = takeU ? ucur : vcur;
    float d = q - prev;  // delta = diff(qs, prepend=0)
    int ii = (i < n - 1) ? i : (n - 1);  // searchsorted + clip
    int jj = (j < n - 1) ? j : (n - 1);
    float dq = (float)(ii - jj) * invn;  // pos[ii]-pos[jj]
    acc = fmaf(d * dq, dq, acc);
    prev = q;
    if (takeU) { ++i; ucur = (i < n) ? Up[i] : FLT_MAX; }
    else       { ++j; vcur = (j < n) ? Vp[j] : FLT_MAX; }
  }
#pragma unroll
  for (int off = 16; off > 0; off >>= 1) acc += __shfl_xor(acc, off, 32);
  if (lane == 0) w2out[p] = acc;
}

// Per-batch reduction of 2048 w2 values. Uses fp32 WMMA (A=data, B=ones),
// fixed accumulation order => deterministic; exercises v_wmma on gfx1250.
__global__ __launch_bounds__(32) void k_batchsum(const float* __restrict__ w2,
                                                 float* __restrict__ out) {
  const int b = blockIdx.x;
  const int lane = threadIdx.x;  // exactly one wave32, EXEC all ones
  const float* basep = w2 + b * (S_DIM * R_DIM);
#ifdef HAVE_WMMA_F32X4
  v8f c = {0.f, 0.f, 0.f, 0.f, 0.f, 0.f, 0.f, 0.f};
  v2f ones;
  ones.x = 1.0f;
  ones.y = 1.0f;
  const int m = lane & 15;
  const int khi = (lane >> 4) << 1;  // lanes 0-15: K={0,1}; lanes 16-31: K={2,3}
  for (int chunk = 0; chunk < (S_DIM * R_DIM) / 64; ++chunk) {
    const float* cb = basep + chunk * 64;  // 16x4 fp32 A tile
    v2f a;
    a.x = cb[m * 4 + khi];
    a.y = cb[m * 4 + khi + 1];
    // D = A*ones + C : every column of D accumulates the row sums of A
    c = __builtin_amdgcn_wmma_f32_16x16x4_f32(false, a, false, ones, (short)0, c,
                                              false, false);
  }
  float tot = c[0] + c[1] + c[2] + c[3] + c[4] + c[5] + c[6] + c[7];
  tot += __shfl_xor(tot, 16, 32);  // combine M=0..7 with M=8..15 halves
  if (lane == 0) out[b] = tot;
#else
  float s = 0.f;
  for (int idx = lane; idx < S_DIM * R_DIM; idx += 32) s += basep[idx];
#pragma unroll
  for (int off = 16; off > 0; off >>= 1) s += __shfl_xor(s, off, 32);
  if (lane == 0) out[b] = s;
#endif
}

extern "C" void kernel_launch(void* const* d_in, const int* in_sizes, int n_in,
                              void* d_out, int out_size, void* d_ws, size_t ws_size,
                              hipStream_t stream) {
  (void)in_sizes; (void)n_in; (void)out_size; (void)ws_size;
  const float* x = (const float*)d_in[0];
  const float* y = (const float*)d_in[1];
  float* out = (float*)d_out;

  // Workspace layout (floats): [0..15] min slot (+pad), then partial sums,
  // per-problem sums, per-problem w2. Total ~1.1 MB.
  float* wsf = (float*)d_ws;
  unsigned* minslot = (unsigned*)wsf;
  float* pu = wsf + 16;
  float* pv = pu + BS_DIM * TSPLIT * R_DIM;
  float* su = pv + BS_DIM * TSPLIT * R_DIM;
  float* sv = su + NPROB;
  float* w2 = sv + NPROB;

  k_init<<<1, 1, 0, stream>>>(minslot);
  k_partial<<<BS_DIM * TSPLIT, 256, 0, stream>>>(x, y, pu, pv, minslot);
  k_sums<<<NPROB / 256, 256, 0, stream>>>(pu, pv, su, sv);
  k_w2<<<NPROB / WPB, 128, 0, stream>>>(x, y, minslot, su, sv, w2);
  k_batchsum<<<B_DIM, 32, 0, stream>>>(w2, out);
}